// GraphCNN_83889301225993
// MI455X (gfx1250) — compile-verified
//
#include <hip/hip_runtime.h>
#include <math.h>

// ---------------------------------------------------------------------------
// GCN forward for MI455X (gfx1250, wave32).
//   GEMMs:   V_WMMA_F32_16X16X4_F32 (full fp32, matches reference numerics),
//            B panel staged in LDS via async global->LDS (ASYNCcnt),
//            ds_load feeds the matrix pipe, 4x B reuse per block.
//   Scatter: wave-per-edge float4 gather + global_atomic_add_f32.
// Roofline: edge aggregation (~21 GB L2/HBM traffic) dominates (~1 ms @
// 23.3 TB/s); fp32 WMMA GEMMs (~111 GFLOP) finish well under that.
// ---------------------------------------------------------------------------

typedef float v2f __attribute__((ext_vector_type(2)));
typedef float v8f __attribute__((ext_vector_type(8)));
typedef int   v4i __attribute__((ext_vector_type(4)));

#define NN 50000
#define NE 1600000
#define KC 64                 // K-chunk staged in LDS
#define LDSB_STRIDE 80        // padded row stride (floats): conflict-free ds_load

#if defined(__has_builtin)
#if __has_builtin(__builtin_amdgcn_global_load_async_to_lds_b128)
#define HAVE_ASYNC_LDS 1
#endif
#if __has_builtin(__builtin_amdgcn_s_wait_asynccnt)
#define HAVE_WAIT_ASYNC 1
#endif
#endif

#if defined(HAVE_ASYNC_LDS)
typedef __attribute__((address_space(1))) v4i* gbl_v4i_ptr;
typedef __attribute__((address_space(3))) v4i* lds_v4i_ptr;
#endif

// ---------------- helpers ----------------
__global__ void zero_f32_k(float* __restrict__ p, int n) {
    int i = blockIdx.x * blockDim.x + threadIdx.x;
    if (i < n) p[i] = 0.0f;
}

__global__ void edge_deg_k(const int* __restrict__ dst, float* __restrict__ deg, int nE) {
    int e = blockIdx.x * blockDim.x + threadIdx.x;
    if (e < nE) atomicAdd(&deg[dst[e]], 1.0f);
}

__global__ void dinv_k(const float* __restrict__ deg, float* __restrict__ dinv, int n) {
    int i = blockIdx.x * blockDim.x + threadIdx.x;
    if (i < n) {
        float d = deg[i] + 1.0f;           // +1 self-loop; deg >= 1 guaranteed
        dinv[i] = rsqrtf(d);
    }
}

// ---------------- fp32 WMMA GEMM:  C[M,N] = (reluA? relu(A) : A)[M,K] * B[K,N] + bias ----
// wave tile: 16M x 64N, block = 4 waves (64M x 64N). M mult of 16, N mult of
// 64, K mult of KC (512/1024 here). Waves with mtile past the end clamp to the
// last tile (duplicate identical writes) so block barriers stay uniform.
__global__ void __launch_bounds__(128)
gemm_wmma_f32_k(const float* __restrict__ A, const float* __restrict__ B,
                const float* __restrict__ bias, float* __restrict__ C,
                int M, int K, int N, int reluA)
{
    __shared__ float ldsB[KC * LDSB_STRIDE];   // 20.5 KB of the 320 KB WGP LDS

    const int tid   = threadIdx.x;
    const int lane  = tid & 31;
    const int wave  = tid >> 5;
    const int MT    = M >> 4;
    int mtile       = blockIdx.x * 4 + wave;   // 16 rows per tile
    if (mtile >= MT) mtile = MT - 1;           // clamp: keep barriers uniform
    const int n0    = blockIdx.y * 64;         // 64-col panel shared by block

    const int mrow  = mtile * 16 + (lane & 15);
    const int khalf = (lane >> 4) * 2;         // lanes 0-15 -> K{0,1}, 16-31 -> K{2,3}
    const int ncol  = lane & 15;

    v8f acc0 = {}, acc1 = {}, acc2 = {}, acc3 = {};

    const float* __restrict__ Arow = A + (size_t)mrow * K;

    for (int k0 = 0; k0 < K; k0 += KC) {
        // ---- cooperative stage of B[k0..k0+63][n0..n0+63] into LDS ----
        // 64 rows x 16 float4 chunks = 1024 copies; 8 per thread.
#pragma unroll
        for (int it = 0; it < 8; ++it) {
            int linear = it * 128 + tid;
            int r  = linear >> 4;
            int c4 = (linear & 15) * 4;
            const float* gp = B + (size_t)(k0 + r) * N + n0 + c4;
            float* lp = &ldsB[r * LDSB_STRIDE + c4];
#if defined(HAVE_ASYNC_LDS)
            __builtin_amdgcn_global_load_async_to_lds_b128(
                (gbl_v4i_ptr)gp, (lds_v4i_ptr)lp, 0, 0);
#else
            *(float4*)lp = *(const float4*)gp;
#endif
        }
#if defined(HAVE_ASYNC_LDS)
#if defined(HAVE_WAIT_ASYNC)
        __builtin_amdgcn_s_wait_asynccnt(0);
#else
        asm volatile("s_wait_asynccnt 0x0" ::: "memory");
#endif
#endif
        __syncthreads();

        // prefetch the next A chunk (near/WGP locality)
        __builtin_prefetch(Arow + k0 + KC, 0, 3);

        // ---- compute over the staged chunk ----
        for (int kk = 0; kk < KC; kk += 4) {
            // A fragment: 16x4, one row per lane, 2 K-values per lane (b64 load)
            v2f a = *(const v2f*)(Arow + k0 + kk + khalf);
            if (reluA) { a.x = fmaxf(a.x, 0.0f); a.y = fmaxf(a.y, 0.0f); }

            const float* __restrict__ Bp0 = &ldsB[(kk + khalf) * LDSB_STRIDE + ncol];
            const float* __restrict__ Bp1 = Bp0 + LDSB_STRIDE;

            v2f b0, b1, b2, b3;
            b0.x = Bp0[ 0]; b0.y = Bp1[ 0];
            b1.x = Bp0[16]; b1.y = Bp1[16];
            b2.x = Bp0[32]; b2.y = Bp1[32];
            b3.x = Bp0[48]; b3.y = Bp1[48];

            acc0 = __builtin_amdgcn_wmma_f32_16x16x4_f32(false, a, false, b0, (short)0, acc0, false, false);
            acc1 = __builtin_amdgcn_wmma_f32_16x16x4_f32(false, a, false, b1, (short)0, acc1, false, false);
            acc2 = __builtin_amdgcn_wmma_f32_16x16x4_f32(false, a, false, b2, (short)0, acc2, false, false);
            acc3 = __builtin_amdgcn_wmma_f32_16x16x4_f32(false, a, false, b3, (short)0, acc3, false, false);
        }
        __syncthreads();   // protect LDS before next chunk overwrite
    }

    // C/D layout: VGPR r -> M = r (lanes 0-15) / r+8 (lanes 16-31); N = lane%16
    const int rbase = mtile * 16 + ((lane >> 4) ? 8 : 0);
    const int cb    = n0 + ncol;
    const float bias0 = bias[cb +  0];
    const float bias1 = bias[cb + 16];
    const float bias2 = bias[cb + 32];
    const float bias3 = bias[cb + 48];
#pragma unroll
    for (int r = 0; r < 8; ++r) {
        float* __restrict__ Crow = C + (size_t)(rbase + r) * N;
        Crow[cb +  0] = acc0[r] + bias0;
        Crow[cb + 16] = acc1[r] + bias1;
        Crow[cb + 32] = acc2[r] + bias2;
        Crow[cb + 48] = acc3[r] + bias3;
    }
}

// ---------------- aggregation ----------------
// self-loop init: out[i,c] = xw[i,c] * dinv[i]^2
__global__ void agg_init_k(const float* __restrict__ xw, const float* __restrict__ dinv,
                           float* __restrict__ out, int n, int C)
{
    long long idx = (long long)blockIdx.x * blockDim.x + threadIdx.x;
    long long tot = (long long)n * C;
    if (idx < tot) {
        int i = (int)(idx / C);
        float dv = dinv[i];
        out[idx] = xw[idx] * dv * dv;
    }
}

// one wave per edge: gather xw[src]*norm (float4), scatter-add into out[dst]
__global__ void __launch_bounds__(256)
agg_edges_k(const float* __restrict__ xw, const int* __restrict__ src,
            const int* __restrict__ dst, const float* __restrict__ dinv,
            float* __restrict__ out, int nE, int C)
{
    int gwave = (int)(((long long)blockIdx.x * blockDim.x + threadIdx.x) >> 5);
    int lane  = threadIdx.x & 31;
    if (gwave >= nE) return;
    int s = src[gwave];
    int d = dst[gwave];
    float norm = dinv[s] * dinv[d];
    const float4* __restrict__ xs = (const float4*)(xw + (size_t)s * C);
    float* __restrict__ od = out + (size_t)d * C;
    for (int c = lane * 4; c < C; c += 128) {
        float4 v = xs[c >> 2];
        atomicAdd(od + c + 0, v.x * norm);
        atomicAdd(od + c + 1, v.y * norm);
        atomicAdd(od + c + 2, v.z * norm);
        atomicAdd(od + c + 3, v.w * norm);
    }
}

// ---------------- FC (128->10) + log_softmax, relu fused on input ----------------
__global__ void fc_logsoftmax_k(const float* __restrict__ h, const float* __restrict__ W,
                                const float* __restrict__ b, float* __restrict__ out, int M)
{
    int i = blockIdx.x * blockDim.x + threadIdx.x;
    if (i >= M) return;
    const float* __restrict__ hi = h + (size_t)i * 128;
    float logit[10];
#pragma unroll
    for (int o = 0; o < 10; ++o) logit[o] = b[o];
    for (int k = 0; k < 128; ++k) {
        float hv = fmaxf(hi[k], 0.0f);
#pragma unroll
        for (int o = 0; o < 10; ++o) logit[o] = fmaf(hv, W[k * 10 + o], logit[o]);
    }
    float m = logit[0];
#pragma unroll
    for (int o = 1; o < 10; ++o) m = fmaxf(m, logit[o]);
    float s = 0.0f;
#pragma unroll
    for (int o = 0; o < 10; ++o) s += expf(logit[o] - m);
    float lse = m + logf(s);
    float* __restrict__ oi = out + (size_t)i * 10;
#pragma unroll
    for (int o = 0; o < 10; ++o) oi[o] = logit[o] - lse;
}

// ---------------- launcher ----------------
extern "C" void kernel_launch(void* const* d_in, const int* in_sizes, int n_in,
                              void* d_out, int out_size, void* d_ws, size_t ws_size,
                              hipStream_t stream) {
    (void)in_sizes; (void)n_in; (void)out_size; (void)ws_size;

    const float* x   = (const float*)d_in[0];
    const int*   ei  = (const int*)d_in[1];
    const float* W1  = (const float*)d_in[2];
    const float* b1  = (const float*)d_in[3];
    const float* W2  = (const float*)d_in[4];
    const float* b2  = (const float*)d_in[5];
    const float* W3  = (const float*)d_in[6];
    const float* b3  = (const float*)d_in[7];
    const float* Wfc = (const float*)d_in[8];
    const float* bfc = (const float*)d_in[9];
    float* out = (float*)d_out;

    const int* src = ei;            // edge_index[0]
    const int* dst = ei + NE;       // edge_index[1]

    // workspace layout (256B aligned): deg, dinv, P/Q ping-pong (50000x1024 f32)
    char* ws = (char*)d_ws;
    const size_t OFF_DEG  = 0;
    const size_t OFF_DINV = 200192;                       // 50000*4 padded
    const size_t OFF_P    = OFF_DINV + 200192;
    const size_t OFF_Q    = OFF_P + (size_t)NN * 1024 * 4;
    float* deg  = (float*)(ws + OFF_DEG);
    float* dinv = (float*)(ws + OFF_DINV);
    float* P    = (float*)(ws + OFF_P);
    float* Q    = (float*)(ws + OFF_Q);

    const int MT = (NN + 15) / 16;           // 3125 M-tiles
    const int GX = (MT + 3) / 4;             // 4 waves / block

    // normalization
    zero_f32_k<<<(NN + 255) / 256, 256, 0, stream>>>(deg, NN);
    edge_deg_k<<<(NE + 255) / 256, 256, 0, stream>>>(dst, deg, NE);
    dinv_k<<<(NN + 255) / 256, 256, 0, stream>>>(deg, dinv, NN);

    const long long ew_threads = (long long)NE * 32;     // one wave per edge
    const int ew_blocks = (int)((ew_threads + 255) / 256);

    // ---- layer 1: x(512) -> 1024 ----
    gemm_wmma_f32_k<<<dim3(GX, 1024 / 64), 128, 0, stream>>>(x, W1, b1, P, NN, 512, 1024, 0);
    {
        long long tot = (long long)NN * 1024;
        agg_init_k<<<(int)((tot + 255) / 256), 256, 0, stream>>>(P, dinv, Q, NN, 1024);
        agg_edges_k<<<ew_blocks, 256, 0, stream>>>(P, src, dst, dinv, Q, NE, 1024);
    }

    // ---- layer 2: relu(Q)(1024) -> 512 ----
    gemm_wmma_f32_k<<<dim3(GX, 512 / 64), 128, 0, stream>>>(Q, W2, b2, P, NN, 1024, 512, 1);
    {
        long long tot = (long long)NN * 512;
        agg_init_k<<<(int)((tot + 255) / 256), 256, 0, stream>>>(P, dinv, Q, NN, 512);
        agg_edges_k<<<ew_blocks, 256, 0, stream>>>(P, src, dst, dinv, Q, NE, 512);
    }

    // ---- layer 3: relu(Q)(512) -> 128 ----
    gemm_wmma_f32_k<<<dim3(GX, 128 / 64), 128, 0, stream>>>(Q, W3, b3, P, NN, 512, 128, 1);
    {
        long long tot = (long long)NN * 128;
        agg_init_k<<<(int)((tot + 255) / 256), 256, 0, stream>>>(P, dinv, Q, NN, 128);
        agg_edges_k<<<ew_blocks, 256, 0, stream>>>(P, src, dst, dinv, Q, NE, 128);
    }

    // ---- FC + log_softmax (relu fused on load) ----
    fc_logsoftmax_k<<<(NN + 255) / 256, 256, 0, stream>>>(Q, Wfc, bfc, out, NN);
}